// LaneNet_26611617366374
// MI455X (gfx1250) — compile-verified
//
#include <hip/hip_runtime.h>

// ---------------- problem constants (match reference) ----------------
#define BATCH       16
#define HGT         288
#define WID         800
#define HW          (HGT * WID)          // 230400 pixels per batch image
#define NLBL        5                    // labels 0..4 (0 = background)
#define DELTA_V     0.5f

#define BLOCKS_PER_B 90                  // 230400 / 2560
#define PIX_PER_BLK  2560                // 256 threads * 10
#define ITERS        10
#define NBLK         (BATCH * BLOCKS_PER_B)   // 1440
#define PART_STRIDE  28                  // 25 sums/counts (+ pad to float4)
#define META_STRIDE  32                  // 20 means + 5 coef + pad
#define BROW         20                  // padded B-stage row stride (dwords)

typedef float v2f __attribute__((ext_vector_type(2)));
typedef float v8f __attribute__((ext_vector_type(8)));

// Wave-private LDS staging needs no workgroup barrier: LDS ops from one wave
// execute in order (ISA 7.1). We only need to stop compiler reordering.
__device__ __forceinline__ void wave_lds_fence() {
    __builtin_amdgcn_fence(__ATOMIC_SEQ_CST, "wavefront");
    __builtin_amdgcn_wave_barrier();
}

__device__ __forceinline__ unsigned lane_ballot(bool p) {
#if __has_builtin(__builtin_amdgcn_ballot_w32)
    return __builtin_amdgcn_ballot_w32(p);
#else
    return (unsigned)__ballot(p);
#endif
}

// =====================================================================
// Pass 1: per-block {label sums[5][4], counts[5]} via V_WMMA_F32_16X16X4_F32.
//
// One WMMA consumes 4 pixels:
//   A[m][k] = (label[pix k] == m)                (one-hot rows = labels)
//   B[k][n] = emb[pix k][n] (n<4), 1.0 (n==4, count column), 0 otherwise
//   D[m][0..3] += sums[m][*],  D[m][4] += count[m]
//
// Layouts (ISA 7.12.2, wave32):
//   A (16x4 f32, 2 VGPRs): Vr, lane L -> M = L&15, K = r + 2*(L>=16)
//   B (4x16 f32, 2 VGPRs): Vr, lane L -> N = L&15, K = r + 2*(L>=16)
//   D (16x16 f32, 8 VGPRs): Vr, lanes 0-15 -> M=r, N=lane
//
// A is built from 5 wave ballots (label bitmasks) -> bit extract per WMMA.
// B columns 4..15 are loop-invariant and staged once.
// =====================================================================
__global__ __launch_bounds__(256)
void lanenet_pass1(const float* __restrict__ emb,
                   const int*   __restrict__ lab,
                   float*       __restrict__ blockPart)
{
    const int blk    = blockIdx.x;
    const int b      = blk / BLOCKS_PER_B;
    const int blkInB = blk % BLOCKS_PER_B;
    const int tid    = threadIdx.x;
    const int wave   = tid >> 5;
    const int lane   = tid & 31;

    const float* embB = emb + (size_t)b * 4 * HW;
    const int*   labB = lab + (size_t)b * HW;

    __shared__ float sB[8][32][BROW];     // [wave][pixel][column], padded
    __shared__ float sWavePart[8][25];

    // invariant columns of the B stage: col4 = 1.0 (counts), col5..15 = 0
#pragma unroll
    for (int c = 4; c < 16; ++c) sB[wave][lane][c] = (c == 4) ? 1.0f : 0.0f;

    const int base    = blkInB * PIX_PER_BLK + wave * (ITERS * 32);
    const int halfSel = (lane >> 4) << 1;   // 0 (lanes 0-15) / 2 (lanes 16-31)
    const int n       = lane & 15;          // my row (A) / column (B) index

    v8f acc = {0.f, 0.f, 0.f, 0.f, 0.f, 0.f, 0.f, 0.f};

    for (int it = 0; it < ITERS; ++it) {
        const int pix = base + it * 32 + lane;

        const float e0 = embB[0 * HW + pix];
        const float e1 = embB[1 * HW + pix];
        const float e2 = embB[2 * HW + pix];
        const float e3 = embB[3 * HW + pix];
        const int   l  = labB[pix];

        // wave-uniform one-hot bitmasks: bit p == (label of pixel p == m)
        const unsigned m0 = lane_ballot(l == 0);
        const unsigned m1 = lane_ballot(l == 1);
        const unsigned m2 = lane_ballot(l == 2);
        const unsigned m3 = lane_ballot(l == 3);
        const unsigned m4 = lane_ballot(l == 4);

        wave_lds_fence();                  // previous iter's reads done (in-order)
        sB[wave][lane][0] = e0;            // 16B-aligned b128 store (BROW=20)
        sB[wave][lane][1] = e1;
        sB[wave][lane][2] = e2;
        sB[wave][lane][3] = e3;
        wave_lds_fence();

        unsigned myMask = (n == 0) ? m0 : (n == 1) ? m1 : (n == 2) ? m2
                        : (n == 3) ? m3 : (n == 4) ? m4 : 0u;
        myMask >>= halfSel;                // align to my K half

#pragma unroll
        for (int j = 0; j < 8; ++j) {
            const int p0 = 4 * j + halfSel;            // pixel for K=r
            v2f A, Bv;
            A[0]  = (float)((myMask >> (4 * j)) & 1u);
            A[1]  = (float)((myMask >> (4 * j + 1)) & 1u);
            Bv[0] = sB[wave][p0][n];                   // unconditional ds_load
            Bv[1] = sB[wave][p0 + 1][n];
            acc = __builtin_amdgcn_wmma_f32_16x16x4_f32(
                false, A, false, Bv, (short)0, acc, false, false);
        }
    }

    // rows m=0..4 live in acc[m]; lanes 0..4 hold the 5 useful columns
    if (lane < 5) {
        sWavePart[wave][ 0 + lane] = acc[0];
        sWavePart[wave][ 5 + lane] = acc[1];
        sWavePart[wave][10 + lane] = acc[2];
        sWavePart[wave][15 + lane] = acc[3];
        sWavePart[wave][20 + lane] = acc[4];
    }
    __syncthreads();
    if (tid < 25) {
        float s = 0.f;
        for (int wv = 0; wv < 8; ++wv) s += sWavePart[wv][tid];
        blockPart[(size_t)blk * PART_STRIDE + tid] = s;   // j = m*5+n
    }
}

// =====================================================================
// Pass 2: per-batch partial reduction -> means + per-label coefficient
//   coef[l] = present(l) ? (1/max(cnt,1)) / max(num_lanes,1) : 0
// =====================================================================
__global__ __launch_bounds__(32)
void lanenet_meta(const float* __restrict__ blockPart,
                  float*       __restrict__ meta)
{
    const int b = blockIdx.x;
    const int j = threadIdx.x;

    __shared__ float red[25];
    if (j < 25) {
        float s = 0.f;
        for (int i = 0; i < BLOCKS_PER_B; ++i)
            s += blockPart[(size_t)(b * BLOCKS_PER_B + i) * PART_STRIDE + j];
        red[j] = s;
    }
    __syncthreads();

    if (j == 0) {
        float cnt[NLBL];
        int numLanes = 0;
        for (int l = 0; l < NLBL; ++l) {
            cnt[l] = red[l * 5 + 4];
            if (l != 0 && cnt[l] > 0.f) numLanes++;
        }
        const float nd = (float)(numLanes > 1 ? numLanes : 1);
        for (int l = 0; l < NLBL; ++l) {
            const float dden = fmaxf(cnt[l], 1.0f);
            for (int d = 0; d < 4; ++d)
                meta[(size_t)b * META_STRIDE + l * 4 + d] = red[l * 5 + d] / dden;
            const bool present = (l != 0) && (cnt[l] > 0.f);
            meta[(size_t)b * META_STRIDE + 20 + l] =
                present ? (1.0f / dden) / nd : 0.0f;
        }
    }
}

// =====================================================================
// Pass 3: per-pixel hinge vs own centroid (weighted by coef[label]) and
// the weighted 2-class cross-entropy:  nll = max(d,0)+log1p(exp(-|d|)).
// =====================================================================
__global__ __launch_bounds__(256)
void lanenet_var_ce(const float* __restrict__ emb,
                    const float* __restrict__ bseg,
                    const int*   __restrict__ lab,
                    const float* __restrict__ meta,
                    float*       __restrict__ outPart)
{
    const int blk    = blockIdx.x;
    const int b      = blk / BLOCKS_PER_B;
    const int blkInB = blk % BLOCKS_PER_B;
    const int tid    = threadIdx.x;

    __shared__ __align__(16) float sMeta[25];   // 20 means + 5 coef
    if (tid < 25) sMeta[tid] = meta[(size_t)b * META_STRIDE + tid];
    __syncthreads();

    const float* embB  = emb  + (size_t)b * 4 * HW;
    const float* bsegB = bseg + (size_t)b * 2 * HW;
    const int*   labB  = lab  + (size_t)b * HW;
    const int    base  = blkInB * PIX_PER_BLK;

    float accv = 0.f, ceN = 0.f, ceD = 0.f;
    for (int it = 0; it < ITERS; ++it) {
        const int pix = base + it * 256 + tid;
        const int l   = labB[pix];

        // variance hinge
        const float4 mu = *(const float4*)&sMeta[l * 4];   // ds_load_b128
        const float d0 = embB[0 * HW + pix] - mu.x;
        const float d1 = embB[1 * HW + pix] - mu.y;
        const float d2 = embB[2 * HW + pix] - mu.z;
        const float d3 = embB[3 * HW + pix] - mu.w;
        const float sq = d0 * d0 + d1 * d1 + d2 * d2 + d3 * d3;
        const float dist = (sq > 0.f) ? sqrtf(sq) : 0.f;   // zero-safe norm
        const float h = fmaxf(dist - DELTA_V, 0.f);
        accv += sMeta[20 + l] * h * h;

        // weighted binary cross-entropy
        const float x0 = bsegB[0 * HW + pix];
        const float x1 = bsegB[1 * HW + pix];
        const int   t  = (l > 0) ? 1 : 0;
        const float dl = t ? (x0 - x1) : (x1 - x0);        // x_other - x_target
        const float nll = fmaxf(dl, 0.f) + __logf(1.f + __expf(-fabsf(dl)));
        const float w = t ? 1.0f : 0.5f;
        ceN += w * nll;
        ceD += w;
    }

    __shared__ float rV[256], rN[256], rD[256];
    rV[tid] = accv; rN[tid] = ceN; rD[tid] = ceD;
    __syncthreads();
    for (int s = 128; s > 0; s >>= 1) {
        if (tid < s) {
            rV[tid] += rV[tid + s];
            rN[tid] += rN[tid + s];
            rD[tid] += rD[tid + s];
        }
        __syncthreads();
    }
    if (tid == 0) {
        outPart[(size_t)blk * 4 + 0] = rV[0];
        outPart[(size_t)blk * 4 + 1] = rN[0];
        outPart[(size_t)blk * 4 + 2] = rD[0];
    }
}

// =====================================================================
// Final: loss = ceN/ceD + var_total/B   (dist_loss == 0, reg_loss == 0)
// =====================================================================
__global__ __launch_bounds__(256)
void lanenet_final(const float* __restrict__ outPart,
                   float*       __restrict__ out)
{
    const int tid = threadIdx.x;
    float v = 0.f, nn = 0.f, dd = 0.f;
    for (int i = tid; i < NBLK; i += 256) {
        v  += outPart[(size_t)i * 4 + 0];
        nn += outPart[(size_t)i * 4 + 1];
        dd += outPart[(size_t)i * 4 + 2];
    }
    __shared__ float rV[256], rN[256], rD[256];
    rV[tid] = v; rN[tid] = nn; rD[tid] = dd;
    __syncthreads();
    for (int s = 128; s > 0; s >>= 1) {
        if (tid < s) {
            rV[tid] += rV[tid + s];
            rN[tid] += rN[tid + s];
            rD[tid] += rD[tid + s];
        }
        __syncthreads();
    }
    if (tid == 0) out[0] = rN[0] / rD[0] + rV[0] / (float)BATCH;
}

// =====================================================================
extern "C" void kernel_launch(void* const* d_in, const int* in_sizes, int n_in,
                              void* d_out, int out_size, void* d_ws, size_t ws_size,
                              hipStream_t stream)
{
    const float* emb  = (const float*)d_in[0];   // [16,4,288,800] f32
    const float* bseg = (const float*)d_in[1];   // [16,2,288,800] f32
    const int*   lab  = (const int*)  d_in[2];   // [16,288,800]   i32
    float*       out  = (float*)d_out;

    float* blockPart = (float*)d_ws;                            // NBLK * 28
    float* meta      = blockPart + (size_t)NBLK * PART_STRIDE;  // 16 * 32
    float* outPart   = meta + (size_t)BATCH * META_STRIDE;      // NBLK * 4

    lanenet_pass1 <<<NBLK, 256, 0, stream>>>(emb, lab, blockPart);
    lanenet_meta  <<<BATCH, 32, 0, stream>>>(blockPart, meta);
    lanenet_var_ce<<<NBLK, 256, 0, stream>>>(emb, bseg, lab, meta, outPart);
    lanenet_final <<<1, 256, 0, stream>>>(outPart, out);
}